// SetCriterion_22943715295840
// MI455X (gfx1250) — compile-verified
//
#include <hip/hip_runtime.h>
#include <math.h>

#define B_ 32
#define Q_ 1000
#define G_ 64

typedef float v2f __attribute__((ext_vector_type(2)));
typedef float v8f __attribute__((ext_vector_type(8)));

__device__ __forceinline__ float giou_f(float ax1, float ay1, float ax2, float ay2,
                                        float bx1, float by1, float bx2, float by2) {
    float ix1 = fmaxf(ax1, bx1), iy1 = fmaxf(ay1, by1);
    float ix2 = fminf(ax2, bx2), iy2 = fminf(ay2, by2);
    float inter = fmaxf(ix2 - ix1, 0.0f) * fmaxf(iy2 - iy1, 0.0f);
    float areaA = fmaxf(ax2 - ax1, 0.0f) * fmaxf(ay2 - ay1, 0.0f);
    float areaB = fmaxf(bx2 - bx1, 0.0f) * fmaxf(by2 - by1, 0.0f);
    float uni = areaA + areaB - inter + 1e-6f;
    float iou = inter / uni;
    float ex1 = fminf(ax1, bx1), ey1 = fminf(ay1, by1);
    float ex2 = fmaxf(ax2, bx2), ey2 = fmaxf(ay2, by2);
    float enc = fmaxf(fmaxf(ex2 - ex1, 0.0f) * fmaxf(ey2 - ey1, 0.0f), 1e-6f);
    return iou - (enc - uni) / enc;
}

// ---------------------------------------------------------------------------
// Kernel 1: cost matrix, stored TRANSPOSED: costT[b][g][q] (64 rows x 1000 cols
// per batch) so the Hungarian row scans are coalesced/contiguous.
// One block per (b,g); threads stride over q.
// ---------------------------------------------------------------------------
__global__ void cost_kernel(const float* __restrict__ pb, const float* __restrict__ pl,
                            const float* __restrict__ gb, float* __restrict__ costT) {
    int bg = blockIdx.x;
    int b = bg >> 6;
    int g = bg & 63;
    const float* gt = gb + (size_t)(b * G_ + g) * 4;
    float gx1 = gt[0], gy1 = gt[1], gx2 = gt[2], gy2 = gt[3];
    float* crow = costT + (size_t)(b * G_ + g) * Q_;
    for (int q = threadIdx.x; q < Q_; q += blockDim.x) {
        const float* p = pb + (size_t)(b * Q_ + q) * 4;
        float px1 = p[0], py1 = p[1], px2 = p[2], py2 = p[3];
        const float* lg = pl + (size_t)(b * Q_ + q) * 2;
        float prob0 = 1.0f / (1.0f + expf(lg[1] - lg[0]));   // softmax[...,0]
        float l1c = fabsf(px1 - gx1) + fabsf(py1 - gy1) + fabsf(px2 - gx2) + fabsf(py2 - gy2);
        float gi = giou_f(px1, py1, px2, py2, gx1, gy1, gx2, gy2);
        crow[q] = 5.0f * l1c - prob0 - 2.0f * gi;            // COST_BBOX*l1 + COST_CLASS*(-p0) + COST_GIOU*(-giou)
    }
}

// ---------------------------------------------------------------------------
// Kernel 2: Jonker-Volgenant Hungarian, one block (8 wave32) per batch.
// n=64 rows (gt), m=1000 cols (pred). All state in LDS, float64 duals to match
// the reference's float64 arithmetic. Parallel relax + tree argmin per step
// with first-index tie-break (matches np.argmin).
// ---------------------------------------------------------------------------
__global__ void hungarian_kernel(const float* __restrict__ costT, int* __restrict__ pred_idx) {
    const int b = blockIdx.x;
    const int tid = threadIdx.x;
    const int T = 256;
    __shared__ double u[G_ + 1];
    __shared__ double v[Q_ + 1];
    __shared__ double minv[Q_ + 1];
    __shared__ int p[Q_ + 1];
    __shared__ int way[Q_ + 1];
    __shared__ unsigned char used[Q_ + 1];
    __shared__ double redv[256];
    __shared__ int redi[256];
    __shared__ int sj0;
    __shared__ double sdelta;

    for (int j = tid; j <= Q_; j += T) { v[j] = 0.0; p[j] = 0; way[j] = 0; }
    for (int j = tid; j <= G_; j += T) u[j] = 0.0;
    __syncthreads();

    const float* cb = costT + (size_t)b * G_ * Q_;
    const double INFD = __builtin_inf();

    for (int i = 1; i <= G_; ++i) {
        if (tid == 0) { p[0] = i; sj0 = 0; }
        for (int j = tid; j <= Q_; j += T) { minv[j] = INFD; used[j] = 0; }
        __syncthreads();
        while (true) {
            int j0 = sj0;
            if (tid == 0) used[j0] = 1;
            __syncthreads();
            int i0 = p[j0];
            const float* row = cb + (size_t)(i0 - 1) * Q_;
            double ui0 = u[i0];
            double bestv = INFD;
            int bestj = 0;
            for (int j = tid + 1; j <= Q_; j += T) {
                if (!used[j]) {
                    double cur = (double)row[j - 1] - ui0 - v[j];
                    if (cur < minv[j]) { minv[j] = cur; way[j] = j0; }
                    double mv = minv[j];
                    if (mv < bestv) { bestv = mv; bestj = j; }  // strict < keeps smallest j
                }
            }
            redv[tid] = bestv;
            redi[tid] = bestj;
            __syncthreads();
            for (int s = T / 2; s > 0; s >>= 1) {
                if (tid < s) {
                    double v2 = redv[tid + s];
                    int i2 = redi[tid + s];
                    if (v2 < redv[tid] || (v2 == redv[tid] && i2 < redi[tid])) {
                        redv[tid] = v2; redi[tid] = i2;
                    }
                }
                __syncthreads();
            }
            if (tid == 0) { sdelta = redv[0]; sj0 = redi[0]; }
            __syncthreads();
            double delta = sdelta;
            for (int j = tid; j <= Q_; j += T) {
                if (used[j]) { u[p[j]] += delta; v[j] -= delta; }  // p[j] distinct over used j
                else         { minv[j] -= delta; }
            }
            __syncthreads();
            if (p[sj0] == 0) break;   // uniform: shared read after barrier
        }
        if (tid == 0) {               // augment along alternating path
            int j0 = sj0;
            while (j0) { int j1 = way[j0]; p[j0] = p[j1]; j0 = j1; }
        }
        __syncthreads();
    }
    for (int j = tid + 1; j <= Q_; j += T) {
        int pj = p[j];
        if (pj > 0) pred_idx[b * G_ + (pj - 1)] = j - 1;   // col4row
    }
}

// ---------------------------------------------------------------------------
// Kernel 3: per-batch loss partials (class NLL normalized per batch, L1, GIoU).
// Deterministic fixed-order LDS tree reductions; no atomics.
// ---------------------------------------------------------------------------
__device__ double block_reduce256(double val, double* buf, int tid) {
    buf[tid] = val;
    __syncthreads();
    for (int s = 128; s > 0; s >>= 1) {
        if (tid < s) buf[tid] += buf[tid + s];
        __syncthreads();
    }
    double r = buf[0];
    __syncthreads();
    return r;
}

__global__ void losses_kernel(const float* __restrict__ pb, const float* __restrict__ pl,
                              const float* __restrict__ gb, const int* __restrict__ pidx,
                              float* __restrict__ partials) {
    const int b = blockIdx.x, tid = threadIdx.x, T = 256;
    __shared__ unsigned char lab[Q_];
    __shared__ double buf[256];
    for (int q = tid; q < Q_; q += T) lab[q] = 1;
    __syncthreads();
    if (tid < G_) lab[pidx[b * G_ + tid]] = 0;
    __syncthreads();
    double wnll = 0.0, wsum = 0.0;
    for (int q = tid; q < Q_; q += T) {
        const float* l = pl + (size_t)(b * Q_ + q) * 2;
        float l0 = l[0], l1 = l[1];
        float m = fmaxf(l0, l1);
        float lse = m + logf(expf(l0 - m) + expf(l1 - m));
        int lb_ = lab[q];
        float lp = (lb_ == 0 ? l0 : l1) - lse;   // log_softmax at label
        float w = (lb_ == 0) ? 1.0f : 0.1f;      // EOS_COEF
        wnll += (double)(w * (-lp));
        wsum += (double)w;
    }
    double rn = block_reduce256(wnll, buf, tid);
    double rw = block_reduce256(wsum, buf, tid);
    double lbb = 0.0, lgg = 0.0;
    if (tid < G_) {
        int pi = pidx[b * G_ + tid];
        const float* p = pb + (size_t)(b * Q_ + pi) * 4;
        const float* g = gb + (size_t)(b * G_ + tid) * 4;
        lbb = (double)(fabsf(p[0] - g[0]) + fabsf(p[1] - g[1]) + fabsf(p[2] - g[2]) + fabsf(p[3] - g[3]));
        lgg = (double)(1.0f - giou_f(p[0], p[1], p[2], p[3], g[0], g[1], g[2], g[3]));
    }
    double rb = block_reduce256(lbb, buf, tid);
    double rg = block_reduce256(lgg, buf, tid);
    if (tid == 0) {
        partials[0 * B_ + b] = (float)(rn / rw);
        partials[1 * B_ + b] = (float)rb;
        partials[2 * B_ + b] = (float)rg;
    }
}

// ---------------------------------------------------------------------------
// Kernel 4: finalize via V_WMMA_F32_16X16X4_F32 (one wave32, EXEC all-ones).
// Pre-pair 32 partials -> 16 per loss; pack as A rows {0-3:class, 4-7:bbox,
// 8-11:giou, 12-15:zero}; B = ones(4x16); D[m][0] = row sums.
// D layout: lane 0 holds M=0..7 (VGPR 0..7) at N=0; lane 16 holds M=8..15.
// ---------------------------------------------------------------------------
__global__ void finalize_kernel(const float* __restrict__ partials, float* __restrict__ out) {
    const int lane = threadIdx.x;  // 0..31, one full wave32
    __shared__ float ys[4][16];
    for (int t = lane; t < 64; t += 32) {
        int s = t >> 4, j = t & 15;
        ys[s][j] = (s < 3) ? (partials[s * B_ + 2 * j] + partials[s * B_ + 2 * j + 1]) : 0.0f;
    }
    __syncthreads();
    // A 16x4 layout: lanes 0-15 -> A[m][0..1] in VGPR0..1; lanes 16-31 -> A[m][2..3]
    int m = lane & 15;
    int s = m >> 2, i = m & 3;
    v2f a;
    if (lane < 16) { a.x = ys[s][4 * i + 0]; a.y = ys[s][4 * i + 1]; }
    else           { a.x = ys[s][4 * i + 2]; a.y = ys[s][4 * i + 3]; }
    v2f bb;
    bb.x = 1.0f;
    bb.y = 1.0f;       // B = ones: layout-invariant
    v8f c = {0.0f, 0.0f, 0.0f, 0.0f, 0.0f, 0.0f, 0.0f, 0.0f};
    v8f d = __builtin_amdgcn_wmma_f32_16x16x4_f32(false, a, false, bb, (short)0, c, false, false);
    __shared__ float tot[3];
    if (lane == 0)  { tot[0] = d[0] + d[1] + d[2] + d[3]; tot[1] = d[4] + d[5] + d[6] + d[7]; }
    if (lane == 16) { tot[2] = d[0] + d[1] + d[2] + d[3]; }
    __syncthreads();
    if (lane == 0) {
        float lc = tot[0] / (float)B_;
        float lb = tot[1] / (float)(B_ * G_);
        float lg = tot[2] / (float)(B_ * G_);
        out[0] = lc + 5.0f * lb + 2.0f * lg;   // LOSS_CLASS/BBOX/GIOU weights
        out[1] = lc;
        out[2] = lb;
        out[3] = lg;
    }
}

extern "C" void kernel_launch(void* const* d_in, const int* in_sizes, int n_in,
                              void* d_out, int out_size, void* d_ws, size_t ws_size,
                              hipStream_t stream) {
    (void)in_sizes; (void)n_in; (void)out_size; (void)ws_size;
    const float* pb = (const float*)d_in[0];   // pred_boxes  [32,1000,4]
    const float* pl = (const float*)d_in[1];   // pred_logits [32,1000,2]
    const float* gb = (const float*)d_in[2];   // gt_boxes    [32,64,4]
    float* out = (float*)d_out;                // [4] f32

    char* ws = (char*)d_ws;
    float* costT    = (float*)ws;                                              // 32*64*1000 f32 (8 MB)
    int*   pidx     = (int*)(ws + (size_t)B_ * G_ * Q_ * sizeof(float));       // 32*64 i32
    float* partials = (float*)(ws + (size_t)B_ * G_ * Q_ * sizeof(float)
                                  + (size_t)B_ * G_ * sizeof(int));            // 3*32 f32

    cost_kernel<<<dim3(B_ * G_), dim3(256), 0, stream>>>(pb, pl, gb, costT);
    hungarian_kernel<<<dim3(B_), dim3(256), 0, stream>>>(costT, pidx);
    losses_kernel<<<dim3(B_), dim3(256), 0, stream>>>(pb, pl, gb, pidx, partials);
    finalize_kernel<<<dim3(1), dim3(32), 0, stream>>>(partials, out);
}